// mbr_area_compute_23639499997219
// MI455X (gfx1250) — compile-verified
//
#include <hip/hip_runtime.h>

// MBR (min bounding rectangle) areas via convex-hull edge test + rotating calipers.
// One wave32 handles 2 samples; the pairwise cross matrix C = P * [y,-x]^T for both
// samples is computed with a single block-diagonal V_WMMA_F32_16X16X4_F32.
//
// C is antisymmetric, so lane i's WMMA column d[.] = C[.][i] is also -row(i).
// After broadcasting column J to the group (8 independent swizzles), the hull-edge
// test for pair (i=kloc, J) is fully lane-local:
//   valid  <=>  min_k (d[k] - r_k) >= d[J],   r_k = C[k][J] (broadcast from lane J)
// Lane i then evaluates the candidate boxes for its pairs (i, J) in registers.

typedef __attribute__((ext_vector_type(2))) float v2f;
typedef __attribute__((ext_vector_type(8))) float v8f;

#define KPTS 8

// ds_swizzle group-of-32 encoding: offset = xor<<10 | or<<5 | and
#define SWZ_XOR1 0x041F
#define SWZ_XOR2 0x081F
#define SWZ_XOR4 0x101F
#define SWZ_BCAST(J) (((J) << 5) | 0x18)   // new_lane = (lane & 0x18) | J

#define SWZF(v, pat) \
  __int_as_float(__builtin_amdgcn_ds_swizzle(__float_as_int(v), (pat)))

// Raw v_min_num_f32 for the swizzle-fed final merge (avoids canonicalize ops).
__device__ __forceinline__ float vminq(float a, float b) {
  float r;
  asm("v_min_num_f32 %0, %1, %2" : "=v"(r) : "v"(a), "v"(b));
  return r;
}

// Hull-edge test for all pairs (*, J): 8 independent broadcasts of column J, then
// a lane-local min-tree. Macro so the swizzle pattern is an integer-constant expr.
#define MASK_ROW(J) {                                                          \
    float h0 = d[0] - SWZF(d[0], SWZ_BCAST(J));                                \
    float h1 = d[1] - SWZF(d[1], SWZ_BCAST(J));                                \
    float h2 = d[2] - SWZF(d[2], SWZ_BCAST(J));                                \
    float h3 = d[3] - SWZF(d[3], SWZ_BCAST(J));                                \
    float h4 = d[4] - SWZF(d[4], SWZ_BCAST(J));                                \
    float h5 = d[5] - SWZF(d[5], SWZ_BCAST(J));                                \
    float h6 = d[6] - SWZF(d[6], SWZ_BCAST(J));                                \
    float h7 = d[7] - SWZF(d[7], SWZ_BCAST(J));                                \
    float h = fminf(fminf(fminf(h0, h1), fminf(h2, h3)),                       \
                    fminf(fminf(h4, h5), fminf(h6, h7)));                      \
    if (h >= d[(J)]) vbits |= (1u << (J));                                     \
  }

__global__ __launch_bounds__(256) void mbr_area_kernel(const float* __restrict__ corners,
                                                       float* __restrict__ out,
                                                       int nbox) {
  const int tid  = blockIdx.x * blockDim.x + threadIdx.x;
  const int lane = threadIdx.x & 31;
  const int wave = tid >> 5;
  if (2 * wave >= nbox) return;                    // wave-uniform: EXEC stays all-ones

  const int sample = 2 * wave + (lane >> 4);       // lanes 0-15 -> sample A, 16-31 -> B
  const float* base = corners + (size_t)sample * (KPTS * 2);

  // Full point set of this lane's sample (compile-time indexed after unroll).
  float px[KPTS], py[KPTS];
#pragma unroll
  for (int q = 0; q < 4; ++q) {
    float4 f = ((const float4*)base)[q];
    px[2 * q]     = f.x;  py[2 * q]     = f.y;
    px[2 * q + 1] = f.z;  py[2 * q + 1] = f.w;
  }

  // Point (lane&7): WMMA operand, this lane's C column index, and the 'i' endpoint
  // of every pair this lane owns.
  const int    kloc = lane & 7;
  const float2 pk   = ((const float2*)base)[kloc];

  // Block-diagonal 2-sample batch:
  //   A (16x4): rows 0-7  = sample A (x, y, 0, 0); rows 8-15 = sample B (0, 0, x, y)
  //   B (4x16): cols 0-7  = sample A (y,-x, 0, 0)^T; cols 8-15 = sample B (0, 0, y,-x)^T
  //   D = diag(C_A, C_B),  C[a][b] = x_a*y_b - y_a*x_b  (= p_a x p_b)
  // f32 16x4 / 4x16 operand layout: VGPR0 = K0 (lanes 0-15) | K2 (lanes 16-31),
  //                                 VGPR1 = K1 (lanes 0-15) | K3 (lanes 16-31).
  const int  m        = lane & 15;
  const bool opActive = ((m < 8) == ((lane >> 4) == 0));   // lanes 0-7 and 24-31
  v2f A, B;
  A.x = opActive ? pk.x : 0.0f;
  A.y = opActive ? pk.y : 0.0f;
  B.x = A.y;          //  y
  B.y = -A.x;         // -x

  v8f acc = {0.f, 0.f, 0.f, 0.f, 0.f, 0.f, 0.f, 0.f};
  v8f d = __builtin_amdgcn_wmma_f32_16x16x4_f32(false, A, false, B, (short)0, acc,
                                                false, false);
  // D layout: VGPR g, lane l -> row g + 8*(l>=16), col l%16.
  //   lanes 0-7   : d[j] = C_A[j][k], k = lane
  //   lanes 24-31 : d[j] = C_B[j][k], k = lane - 24
  //   lanes 8-23  : zeros (off-diagonal blocks) -- their results never escape
  //                 their own 8-lane swizzle group and are never stored.

  // ---- Stage 1: hull-edge bitmask; lane kloc gets bit J for pair (kloc, J). ----
  unsigned vbits = 0u;
  MASK_ROW(0) MASK_ROW(1) MASK_ROW(2) MASK_ROW(3)
  MASK_ROW(4) MASK_ROW(5) MASK_ROW(6) MASK_ROW(7)
  vbits &= ~(1u << kloc);                          // drop the vacuous self-pair

  // ---- Stage 2: per-lane rotating calipers for this lane's pairs (kloc, J). ----
  float best = __builtin_huge_valf();
#pragma unroll
  for (int J = 0; J < 8; ++J) {
    // Exact algebraic fold of cos(|fmod(atan2(dy,dx), pi/2)|):
    //   (c,s) = u>=0 ? (u,|v|) : (|v|,-u),  (u,v) = (dx,dy)/||d||
    float dx = px[J] - pk.x;
    float dy = py[J] - pk.y;
    float l2 = dx * dx + dy * dy;
    float iv = (l2 > 0.0f) ? rsqrtf(l2) : 0.0f;
    float u  = dx * iv, v = dy * iv;
    float av = fabsf(v);
    float c  = (u >= 0.0f) ? u  : av;
    float s  = (u >= 0.0f) ? av : -u;
    if (!(l2 > 0.0f)) { c = 1.0f; s = 0.0f; }      // degenerate duplicate points
    // Rotate all 8 points of this sample in registers; running min/max (canonical
    // VALU inputs -> plain v_min_num/v_max_num, no DS traffic, no waits).
    float rx0 = c * px[0] + s * py[0], ry0 = c * py[0] - s * px[0];
    float mnx = rx0, mxx = rx0, mny = ry0, mxy = ry0;
#pragma unroll
    for (int t = 1; t < 8; ++t) {
      float rx = c * px[t] + s * py[t];
      float ry = c * py[t] - s * px[t];
      mnx = fminf(mnx, rx); mxx = fmaxf(mxx, rx);
      mny = fminf(mny, ry); mxy = fmaxf(mxy, ry);
    }
    float area = (mxx - mnx) * (mxy - mny);
    if (vbits & (1u << J)) best = fminf(best, area);
  }

  // ---- Stage 3: merge the 8 lanes of each sample group. ----
  best = vminq(best, SWZF(best, SWZ_XOR1));
  best = vminq(best, SWZF(best, SWZ_XOR2));
  best = vminq(best, SWZF(best, SWZ_XOR4));

  if (lane == 0)  out[sample] = best;              // sample = 2*wave
  if (lane == 24) out[sample] = best;              // sample = 2*wave + 1
}

extern "C" void kernel_launch(void* const* d_in, const int* in_sizes, int n_in,
                              void* d_out, int out_size, void* d_ws, size_t ws_size,
                              hipStream_t stream) {
  (void)n_in; (void)d_ws; (void)ws_size; (void)out_size;
  const float* corners = (const float*)d_in[0];
  float*       out     = (float*)d_out;
  const int nbox    = in_sizes[0] / (KPTS * 2);    // 65536
  const int waves   = (nbox + 1) / 2;              // 2 samples per wave32
  const int threads = 256;
  const int blocks  = (waves * 32 + threads - 1) / threads;
  mbr_area_kernel<<<blocks, threads, 0, stream>>>(corners, out, nbox);
}